// SparseNodeAggregator_8126078124632
// MI455X (gfx1250) — compile-verified
//
#include <hip/hip_runtime.h>
#include <hip/hip_bf16.h>

// Problem constants (from reference)
constexpr int Bc = 8, Nc = 20000, Cc = 128, Kc = 64, Ec = 320000;
#define XS   132   // padded LDS stride for 128-wide x tiles (bank-conflict avoidance)
#define HSTR 68    // padded LDS stride for 64-wide tiles

typedef __attribute__((ext_vector_type(2))) float v2f;
typedef __attribute__((ext_vector_type(8))) float v8f;

__device__ __forceinline__ v8f wmma_f32(v2f a, v2f b, v8f c) {
  // V_WMMA_F32_16X16X4_F32 : D(16x16) = A(16x4) * B(4x16) + C
  return __builtin_amdgcn_wmma_f32_16x16x4_f32(false, a, false, b, (short)0, c, false, false);
}

__device__ __forceinline__ void lds_fence() {
  // order cross-lane LDS producer->consumer within a wave (CDNA5 split counters)
  asm volatile("s_wait_dscnt 0" ::: "memory");
}

// ---- CDNA5 async global<->LDS path (ASYNCcnt) ------------------------------
typedef __attribute__((address_space(3))) float lds_f;

__device__ __forceinline__ unsigned lds_off(const float* p) {
  return (unsigned)(unsigned long long)(lds_f*)p;
}
__device__ __forceinline__ void async_ld_b128(const float* lds_dst, const float* gsrc) {
  asm volatile("global_load_async_to_lds_b128 %0, %1, off"
               :: "v"(lds_off(lds_dst)), "v"(gsrc) : "memory");
}
__device__ __forceinline__ void async_ld_b64(const float* lds_dst, const float* gsrc) {
  asm volatile("global_load_async_to_lds_b64 %0, %1, off"
               :: "v"(lds_off(lds_dst)), "v"(gsrc) : "memory");
}
__device__ __forceinline__ void async_st_b128(float* gdst, const float* lds_src) {
  asm volatile("global_store_async_from_lds_b128 %0, %1, off"
               :: "v"(gdst), "v"(lds_off(lds_src)) : "memory");
}
template <int N>
__device__ __forceinline__ void wait_async() {
  asm volatile("s_wait_asynccnt %0" :: "i"(N) : "memory");
}

// ---------------------------------------------------------------------------
// Kernel 0: zero the atomic accumulation regions of d_out
// ---------------------------------------------------------------------------
__global__ void k_zero(float* __restrict__ p, int n) {
  int i = blockIdx.x * blockDim.x + threadIdx.x;
  if (i < n) p[i] = 0.0f;
}

// ---------------------------------------------------------------------------
// Kernel 1: per-node  h = relu(x@W1+b1); S = softmax(h@W2+b2)*m  -> Sg (ws)
// One wave handles a 16-node tile with f32 WMMA; x staged via async-to-LDS,
// S written back with async-from-LDS (drains under next tile's x staging).
// ---------------------------------------------------------------------------
__global__ void __launch_bounds__(256) k_node(
    const float* __restrict__ x, const int* __restrict__ mask,
    const float* __restrict__ W1, const float* __restrict__ b1,
    const float* __restrict__ W2, const float* __restrict__ b2,
    float* __restrict__ Sg)
{
  extern __shared__ float smf[];
  const int b    = blockIdx.y;
  const int tid  = threadIdx.x;
  const int lane = tid & 31;
  const int wave = tid >> 5;
  const int nwb  = blockDim.x >> 5;

  float* W1s = smf;                      // 128 x HSTR
  float* W2s = W1s + 128 * HSTR;         //  64 x HSTR
  float* wb  = W2s + 64 * HSTR + wave * (16 * XS + 16 * HSTR + 16);
  float* xs  = wb;                       // 16 x XS   (x tile)
  float* hs  = xs + 16 * XS;             // 16 x HSTR (h / logits / S tile)
  float* msk = hs + 16 * HSTR;           // 16

  for (int i = tid; i < 128 * 64; i += blockDim.x) W1s[(i >> 6) * HSTR + (i & 63)] = W1[i];
  for (int i = tid; i < 64 * 64;  i += blockDim.x) W2s[(i >> 6) * HSTR + (i & 63)] = W2[i];
  __syncthreads();

  const int m16 = lane & 15, eh = lane >> 4;
  float b1v[4], b2v[4];
  for (int ct = 0; ct < 4; ++ct) { b1v[ct] = b1[16 * ct + m16]; b2v[ct] = b2[16 * ct + m16]; }

  const float* xb = x  + (size_t)b * Nc * Cc;
  const int*   mb = mask + (size_t)b * Nc;
  float*       Sb = Sg + (size_t)b * Nc * Kc;

  const int wid = blockIdx.x * nwb + wave;
  const int nw  = gridDim.x * nwb;
  const int ntiles = Nc / 16;

  for (int t = wid; t < ntiles; t += nw) {
    const int n0 = t * 16;
    // stage x tile (16x128) into LDS via async copies (16 x b128)
    for (int i = 0; i < 16; ++i) {
      int f = (i * 32 + lane) * 4, row = f >> 7, col = f & 127;
      async_ld_b128(xs + row * XS + col, xb + (size_t)(n0 + row) * Cc + col);
    }
    if (lane < 16) msk[lane] = (mb[n0 + lane] > 0) ? 1.0f : 0.0f;
    wait_async<0>();   // x tile ready; also drains prior iteration's S store
    lds_fence();

    // h = relu(x@W1 + b1) : 32 k-steps x 4 col tiles
    v8f acc[4];
    for (int ct = 0; ct < 4; ++ct) { v8f z; for (int r = 0; r < 8; ++r) z[r] = b1v[ct]; acc[ct] = z; }
    for (int kk = 0; kk < 32; ++kk) {
      int kc = 4 * kk + 2 * eh;
      v2f a; a[0] = xs[m16 * XS + kc]; a[1] = xs[m16 * XS + kc + 1];
      for (int ct = 0; ct < 4; ++ct) {
        v2f bb; bb[0] = W1s[kc * HSTR + 16 * ct + m16]; bb[1] = W1s[(kc + 1) * HSTR + 16 * ct + m16];
        acc[ct] = wmma_f32(a, bb, acc[ct]);
      }
    }
    for (int ct = 0; ct < 4; ++ct)
      for (int r = 0; r < 8; ++r) {
        float v = acc[ct][r]; v = v > 0.0f ? v : 0.0f;
        hs[(r + 8 * eh) * HSTR + 16 * ct + m16] = v;
      }
    lds_fence();

    // logits = h@W2 + b2 : 16 k-steps x 4 col tiles
    v8f ac2[4];
    for (int ct = 0; ct < 4; ++ct) { v8f z; for (int r = 0; r < 8; ++r) z[r] = b2v[ct]; ac2[ct] = z; }
    for (int kk = 0; kk < 16; ++kk) {
      int kc = 4 * kk + 2 * eh;
      v2f a; a[0] = hs[m16 * HSTR + kc]; a[1] = hs[m16 * HSTR + kc + 1];
      for (int ct = 0; ct < 4; ++ct) {
        v2f bb; bb[0] = W2s[kc * HSTR + 16 * ct + m16]; bb[1] = W2s[(kc + 1) * HSTR + 16 * ct + m16];
        ac2[ct] = wmma_f32(a, bb, ac2[ct]);
      }
    }
    lds_fence();
    for (int ct = 0; ct < 4; ++ct)
      for (int r = 0; r < 8; ++r)
        hs[(r + 8 * eh) * HSTR + 16 * ct + m16] = ac2[ct][r];
    lds_fence();

    // masked softmax per row (lanes 0..15, one row each)
    if (lane < 16) {
      float mx = -3.4e38f;
      for (int j = 0; j < 64; ++j) mx = fmaxf(mx, hs[lane * HSTR + j]);
      float s = 0.0f;
      for (int j = 0; j < 64; ++j) { float e = __expf(hs[lane * HSTR + j] - mx); hs[lane * HSTR + j] = e; s += e; }
      float inv = msk[lane] / s;
      for (int j = 0; j < 64; ++j) hs[lane * HSTR + j] *= inv;
    }
    lds_fence();

    // write S tile to global workspace asynchronously (LDS -> global, 8 x b128)
    for (int i = 0; i < 8; ++i) {
      int f = (i * 32 + lane) * 4, row = f >> 6, col = f & 63;
      async_st_b128(Sb + (size_t)(n0 + row) * Kc + col, hs + row * HSTR + col);
    }
  }
  // S_ENDPGM performs an implicit wait-idle, draining outstanding async stores.
}

// ---------------------------------------------------------------------------
// Kernel 2: pfeat[b] = S^T @ x  (64x128) per-wave accumulator tile,
// double-buffered async-to-LDS staging: next tile's 24 copies fly under WMMA.
// ---------------------------------------------------------------------------
__global__ void __launch_bounds__(256) k_pfeat(
    const float* __restrict__ x, const float* __restrict__ Sg,
    float* __restrict__ pfeat)
{
  extern __shared__ float smf[];
  const int b = blockIdx.y, tid = threadIdx.x, lane = tid & 31, wave = tid >> 5;
  const int nwb = blockDim.x >> 5;
  float* wbase = smf + wave * 2 * (16 * XS + 16 * HSTR);
  float* xsb[2] = { wbase, wbase + (16 * XS + 16 * HSTR) };
  float* ssb[2] = { xsb[0] + 16 * XS, xsb[1] + 16 * XS };

  const float* xb = x  + (size_t)b * Nc * Cc;
  const float* Sb = Sg + (size_t)b * Nc * Kc;
  float* out = pfeat + (size_t)b * Kc * Cc;

  const int m16 = lane & 15, eh = lane >> 4;
  v8f pf[4][8];
  for (int mt = 0; mt < 4; ++mt)
    for (int ct = 0; ct < 8; ++ct) { v8f z; for (int r = 0; r < 8; ++r) z[r] = 0.0f; pf[mt][ct] = z; }

  auto issue = [&](int t, float* xs, float* ss) {
    const int n0 = t * 16;
    for (int i = 0; i < 16; ++i) {
      int f = (i * 32 + lane) * 4, row = f >> 7, col = f & 127;
      async_ld_b128(xs + row * XS + col, xb + (size_t)(n0 + row) * Cc + col);
    }
    for (int i = 0; i < 8; ++i) {
      int f = (i * 32 + lane) * 4, row = f >> 6, col = f & 63;
      async_ld_b128(ss + row * HSTR + col, Sb + (size_t)(n0 + row) * Kc + col);
    }
  };

  const int wid = blockIdx.x * nwb + wave;
  const int nw  = gridDim.x * nwb;
  const int NT  = Nc / 16;

  int cur = 0;
  int t = wid;
  if (t < NT) issue(t, xsb[0], ssb[0]);
  for (; t < NT; t += nw) {
    const int tn = t + nw;
    if (tn < NT) { issue(tn, xsb[cur ^ 1], ssb[cur ^ 1]); wait_async<24>(); }
    else         { wait_async<0>(); }
    float* xs = xsb[cur];
    float* ss = ssb[cur];
    for (int st = 0; st < 4; ++st) {
      int e0 = 4 * st + 2 * eh;
      v2f afr[4], bfr[8];
      for (int mt = 0; mt < 4; ++mt) { afr[mt][0] = ss[e0 * HSTR + 16 * mt + m16]; afr[mt][1] = ss[(e0 + 1) * HSTR + 16 * mt + m16]; }
      for (int ct = 0; ct < 8; ++ct) { bfr[ct][0] = xs[e0 * XS + 16 * ct + m16]; bfr[ct][1] = xs[(e0 + 1) * XS + 16 * ct + m16]; }
      for (int mt = 0; mt < 4; ++mt)
        for (int ct = 0; ct < 8; ++ct)
          pf[mt][ct] = wmma_f32(afr[mt], bfr[ct], pf[mt][ct]);
    }
    cur ^= 1;
  }
  for (int mt = 0; mt < 4; ++mt)
    for (int ct = 0; ct < 8; ++ct)
      for (int r = 0; r < 8; ++r)
        atomicAdd(out + (16 * mt + r + 8 * eh) * Cc + 16 * ct + m16, pf[mt][ct][r]);
}

// ---------------------------------------------------------------------------
// Kernel 3: pooled_adj[b] = sum_e w_e * outer(S[v_e], S[u_e])  via WMMA:
// 16-edge groups -> P=S[v] (async gather to LDS), Q=w*S[u] (register path,
// overlapped with P's async copies); padj += P^T @ Q.
// ---------------------------------------------------------------------------
__global__ void __launch_bounds__(256) k_edge(
    const float* __restrict__ Sg, const int* __restrict__ ei,
    const float* __restrict__ ew, const int* __restrict__ mask,
    float* __restrict__ padj)
{
  extern __shared__ float smf[];
  const int b = blockIdx.y, tid = threadIdx.x, lane = tid & 31, wave = tid >> 5;
  const int nwb = blockDim.x >> 5;
  float* P  = smf + wave * (2 * 16 * HSTR + 48);
  float* Q  = P + 16 * HSTR;
  float* wv = Q + 16 * HSTR;
  int* iu = (int*)(wv + 16);
  int* iv = iu + 16;

  const int*   eib = ei + (size_t)b * 2 * Ec;
  const float* ewb = ew + (size_t)b * Ec;
  const int*   mb  = mask + (size_t)b * Nc;
  const float* Sb  = Sg + (size_t)b * Nc * Kc;
  float* out = padj + (size_t)b * Kc * Kc;

  const int m16 = lane & 15, eh = lane >> 4;
  v8f pa[4][4];
  for (int rt = 0; rt < 4; ++rt)
    for (int ct = 0; ct < 4; ++ct) { v8f z; for (int r = 0; r < 8; ++r) z[r] = 0.0f; pa[rt][ct] = z; }

  const int wid = blockIdx.x * nwb + wave;
  const int nw  = gridDim.x * nwb;
  const int ngroups = Ec / 16;
  for (int g = wid; g < ngroups; g += nw) {
    const int e0 = g * 16;
    if (lane < 16) {
      int u = eib[e0 + lane], v = eib[Ec + e0 + lane];
      float mu = (mb[u] > 0) ? 1.0f : 0.0f;
      float mv = (mb[v] > 0) ? 1.0f : 0.0f;
      iu[lane] = u; iv[lane] = v; wv[lane] = ewb[e0 + lane] * mu * mv;
    }
    lds_fence();
    // P rows: pure copy S[v_row] -> LDS, async b64 per lane (overlaps with Q)
    for (int i = 0; i < 16; ++i) {
      int c = lane * 2;
      int vr = iv[i];
      async_ld_b64(P + i * HSTR + c, Sb + (size_t)vr * Kc + c);
    }
    // Q rows need the w scale -> register path
    for (int i = 0; i < 16; ++i) {
      int c = lane * 2;
      int ur = iu[i]; float wr = wv[i];
      float2 su = *(const float2*)(Sb + (size_t)ur * Kc + c);
      Q[i * HSTR + c]     = wr * su.x;
      Q[i * HSTR + c + 1] = wr * su.y;
    }
    wait_async<0>();
    lds_fence();
    for (int st = 0; st < 4; ++st) {
      int r0 = 4 * st + 2 * eh;
      v2f afr[4], bfr[4];
      for (int rt = 0; rt < 4; ++rt) { afr[rt][0] = P[r0 * HSTR + 16 * rt + m16]; afr[rt][1] = P[(r0 + 1) * HSTR + 16 * rt + m16]; }
      for (int ct = 0; ct < 4; ++ct) { bfr[ct][0] = Q[r0 * HSTR + 16 * ct + m16]; bfr[ct][1] = Q[(r0 + 1) * HSTR + 16 * ct + m16]; }
      for (int rt = 0; rt < 4; ++rt)
        for (int ct = 0; ct < 4; ++ct)
          pa[rt][ct] = wmma_f32(afr[rt], bfr[ct], pa[rt][ct]);
    }
    lds_fence();
  }
  for (int rt = 0; rt < 4; ++rt)
    for (int ct = 0; ct < 4; ++ct)
      for (int r = 0; r < 8; ++r)
        atomicAdd(out + (16 * rt + r + 8 * eh) * Kc + 16 * ct + m16, pa[rt][ct][r]);
}

// ---------------------------------------------------------------------------
// Kernel 4: pmask[b,k] = (sum_n m[b,n] > 0) ? 1 : 0
// ---------------------------------------------------------------------------
__global__ void k_pmask(const int* __restrict__ mask, float* __restrict__ pm) {
  __shared__ float red[256];
  const int b = blockIdx.x, tid = threadIdx.x;
  float s = 0.0f;
  for (int n = tid; n < Nc; n += blockDim.x) s += (mask[(size_t)b * Nc + n] > 0) ? 1.0f : 0.0f;
  red[tid] = s; __syncthreads();
  for (int off = 128; off > 0; off >>= 1) {
    if (tid < off) red[tid] += red[tid + off];
    __syncthreads();
  }
  float val = (red[0] > 0.0f) ? 1.0f : 0.0f;
  if (tid < Kc) pm[b * Kc + tid] = val;
}

// ---------------------------------------------------------------------------
extern "C" void kernel_launch(void* const* d_in, const int* in_sizes, int n_in,
                              void* d_out, int out_size, void* d_ws, size_t ws_size,
                              hipStream_t stream) {
  (void)in_sizes; (void)n_in; (void)out_size; (void)ws_size;
  const float* x    = (const float*)d_in[0];
  const int*   ei   = (const int*)d_in[1];
  const float* ew   = (const float*)d_in[2];
  const int*   mask = (const int*)d_in[3];
  const float* W1   = (const float*)d_in[4];
  const float* b1   = (const float*)d_in[5];
  const float* W2   = (const float*)d_in[6];
  const float* b2   = (const float*)d_in[7];

  float* out   = (float*)d_out;
  float* pfeat = out;                         // 8*64*128
  float* padj  = out + Bc * Kc * Cc;          // 8*64*64
  float* pm    = padj + Bc * Kc * Kc;         // 8*64
  float* Sg    = (float*)d_ws;                // B*N*K fp32 = 40.96 MB

  const int tot = Bc * Kc * Cc + Bc * Kc * Kc + Bc * Kc;
  k_zero<<<dim3((tot + 255) / 256), dim3(256), 0, stream>>>(out, tot);

  size_t smem1 = (size_t)(128 * HSTR + 64 * HSTR + 8 * (16 * XS + 16 * HSTR + 16)) * 4;
  k_node<<<dim3(32, Bc), dim3(256), smem1, stream>>>(x, mask, W1, b1, W2, b2, Sg);

  size_t smem2 = (size_t)(8 * 2 * (16 * XS + 16 * HSTR)) * 4;
  k_pfeat<<<dim3(8, Bc), dim3(256), smem2, stream>>>(x, Sg, pfeat);

  size_t smem3 = (size_t)(8 * (2 * 16 * HSTR + 48)) * 4;
  k_edge<<<dim3(32, Bc), dim3(256), smem3, stream>>>(Sg, ei, ew, mask, padj);

  k_pmask<<<dim3(Bc), dim3(256), 0, stream>>>(mask, pm);
}